// PCTConv_35974646071642
// MI455X (gfx1250) — compile-verified
//
#include <hip/hip_runtime.h>
#include <hip/hip_bf16.h>
#include <math.h>

#define C_TYPES 8
#define N_PPI 10000
#define T_TIS 4
#define N_MG 12
#define IN_CH 128
#define OUT_CH 64
#define HEADS 3
#define DD 192            // OUT_CH*HEADS
#define M_PPI 3
#define E_PPI 50000
#define ETOT_PPI (E_PPI + N_PPI)   // edges + self loops = 60000
#define M_MG 2
#define E_MG 40
#define K_NB 3
#define SEM 128
#define PC 128
#define TISSUE_UPDATE 100

typedef __attribute__((ext_vector_type(2))) float v2f;
typedef __attribute__((ext_vector_type(8))) float v8f;

__device__ __forceinline__ float lrelu(float x, float s) { return x > 0.f ? x : s * x; }

// order-preserving float<->uint encoding for atomicMax-based segment max
__device__ __forceinline__ unsigned f2o(float f) {
    unsigned b = __float_as_uint(f);
    return (b & 0x80000000u) ? ~b : (b | 0x80000000u);
}
__device__ __forceinline__ float o2f(unsigned o) {
    return __uint_as_float((o & 0x80000000u) ? (o & 0x7fffffffu) : ~o);
}
__device__ __forceinline__ void atomAddF(float* p, float v) {
    __hip_atomic_fetch_add(p, v, __ATOMIC_RELAXED, __HIP_MEMORY_SCOPE_AGENT);
}

// ---------------------------------------------------------------------------
// fp32 WMMA GEMM, C[M x N] = A[M x K] * B[K x N], row-major.
// Template: KK = K (128 or 192), TN = N/16 (12 or 8).
// Block = 256 threads = 8 waves; each wave owns a 16-row strip and all TN
// column tiles (TN v8f accumulators). B is staged 64 K-rows at a time into
// LDS with GLOBAL_LOAD_ASYNC_TO_LDS_B128 (ASYNCcnt) so each A fragment load
// feeds TN WMMAs and B fragments come from DS.
// Fragment layouts per CDNA5 ISA 7.12.2 (l = lane&15, h = lane>>4):
//   A v[i] = A[m0+l][k+2h+i]   B v[i] = B[k+2h+i][n0+l]   C v[j] = C[m0+j+8h][n0+l]
// ---------------------------------------------------------------------------
template <int KK, int TN>
__global__ void k_wmma_gemm_t(const float* __restrict__ A, const float* __restrict__ B,
                              float* __restrict__ C, int mtiles)
{
    constexpr int NN = TN * 16;
    constexpr int KC = 64;                      // K-rows of B staged per LDS slab
    __shared__ float sB[KC * NN];               // <= 48 KB
    int tid = threadIdx.x;
    int wave = tid >> 5, lane = tid & 31;
    int strip = blockIdx.x * 8 + wave;          // 16-row strip index
    bool valid = strip < mtiles;
    int l = lane & 15, h = lane >> 4;

    const float* Ar = A + (size_t)((valid ? strip : 0) * 16 + l) * KK;
    v8f zero = {};
    v8f acc[TN];
#pragma unroll
    for (int t = 0; t < TN; ++t) acc[t] = zero;

    unsigned ldsBase = (unsigned)(size_t)&sB[0];

    for (int kc = 0; kc < KK; kc += KC) {
        // --- async-stage B[kc..kc+KC) x NN into LDS, 16B per lane per iter ---
        const float* gB = B + (size_t)kc * NN;
        for (int i = tid; i < (KC * NN) / 4; i += 256) {
            const float* gp = gB + (size_t)i * 4;
            unsigned lo = ldsBase + (unsigned)i * 16u;
            asm volatile("global_load_async_to_lds_b128 %0, %1, off"
                         :: "v"(lo), "v"(gp) : "memory");
        }
        asm volatile("s_wait_asynccnt 0x0" ::: "memory");
        __syncthreads();
        if (valid) {
            for (int k = 0; k < KC; k += 4) {
                v2f a = *(const v2f*)(Ar + kc + k + 2 * h);
#pragma unroll
                for (int t = 0; t < TN; ++t) {
                    v2f b;
                    b.x = sB[(k + 2 * h) * NN + t * 16 + l];
                    b.y = sB[(k + 2 * h + 1) * NN + t * 16 + l];
                    acc[t] = __builtin_amdgcn_wmma_f32_16x16x4_f32(
                        false, a, false, b, (short)0, acc[t], false, false);
                }
            }
        }
        __syncthreads();
    }
    if (valid) {
        float* Cp = C + (size_t)(strip * 16) * NN + l;
#pragma unroll
        for (int t = 0; t < TN; ++t)
#pragma unroll
            for (int j = 0; j < 8; ++j)
                Cp[(size_t)(j + 8 * h) * NN + t * 16] = acc[t][j];
    }
}

// ---------------------------------------------------------------------------
// Edge-softmax GATv2 phase kernels (per celltype; grid covers 3 metapaths)
// ---------------------------------------------------------------------------
__global__ void k_init_edgebufs(unsigned* __restrict__ mmax, float* __restrict__ ssum, int n)
{
    int i = blockIdx.x * blockDim.x + threadIdx.x;
    if (i < n) { mmax[i] = 0x007FFFFFu; /* enc(-inf) */ ssum[i] = 0.f; }
}

__global__ void k_init_agg(float* __restrict__ agg, const float* __restrict__ bias, int total)
{
    int i = blockIdx.x * blockDim.x + threadIdx.x;
    if (i < total) agg[i] = bias[i % DD];
}

__global__ void k_edge_logits(const float* __restrict__ xl, const float* __restrict__ xr,
                              const int* __restrict__ edges,   // [M_PPI][2][E_PPI]
                              const float* __restrict__ att,   // [192] = [3][64]
                              float* __restrict__ elog,        // [M_PPI][ETOT][3]
                              unsigned* __restrict__ mmax)     // [M_PPI][N][3]
{
    int i = blockIdx.x * blockDim.x + threadIdx.x;
    if (i >= M_PPI * ETOT_PPI) return;
    int m = i / ETOT_PPI, e = i % ETOT_PPI;
    int src, dst;
    if (e < E_PPI) { src = edges[(m * 2 + 0) * E_PPI + e]; dst = edges[(m * 2 + 1) * E_PPI + e]; }
    else           { src = dst = e - E_PPI; }
    const float* xs = xl + (size_t)src * DD;
    const float* xd = xr + (size_t)dst * DD;
#pragma unroll
    for (int h = 0; h < HEADS; ++h) {
        float acc = 0.f;
        for (int c = 0; c < OUT_CH; ++c) {
            float v = xs[h * OUT_CH + c] + xd[h * OUT_CH + c];
            v = v > 0.f ? v : 0.2f * v;
            acc += v * att[h * OUT_CH + c];
        }
        elog[(size_t)i * 3 + h] = acc;
        atomicMax(&mmax[(size_t)(m * N_PPI + dst) * 3 + h], f2o(acc));
    }
}

__global__ void k_edge_norm(const int* __restrict__ edges, float* __restrict__ elog,
                            const unsigned* __restrict__ mmax, float* __restrict__ ssum)
{
    int i = blockIdx.x * blockDim.x + threadIdx.x;
    if (i >= M_PPI * ETOT_PPI) return;
    int m = i / ETOT_PPI, e = i % ETOT_PPI;
    int dst = (e < E_PPI) ? edges[(m * 2 + 1) * E_PPI + e] : e - E_PPI;
#pragma unroll
    for (int h = 0; h < HEADS; ++h) {
        float mx = o2f(mmax[(size_t)(m * N_PPI + dst) * 3 + h]);
        float a = __expf(elog[(size_t)i * 3 + h] - mx);
        elog[(size_t)i * 3 + h] = a;
        atomAddF(&ssum[(size_t)(m * N_PPI + dst) * 3 + h], a);
    }
}

__global__ void k_edge_agg(const float* __restrict__ xl, const int* __restrict__ edges,
                           const float* __restrict__ elog, const float* __restrict__ ssum,
                           float* __restrict__ agg)          // [M_PPI][N][192]
{
    int wid = (blockIdx.x * blockDim.x + threadIdx.x) >> 5;  // one wave per edge
    int lane = threadIdx.x & 31;
    if (wid >= M_PPI * ETOT_PPI) return;
    int m = wid / ETOT_PPI, e = wid % ETOT_PPI;
    int src, dst;
    if (e < E_PPI) { src = edges[(m * 2 + 0) * E_PPI + e]; dst = edges[(m * 2 + 1) * E_PPI + e]; }
    else           { src = dst = e - E_PPI; }
    float al[HEADS];
#pragma unroll
    for (int h = 0; h < HEADS; ++h)
        al[h] = elog[(size_t)wid * 3 + h] / (ssum[(size_t)(m * N_PPI + dst) * 3 + h] + 1e-16f);
    const float* xs = xl + (size_t)src * DD;
    float* op = agg + ((size_t)m * N_PPI + dst) * DD;
#pragma unroll
    for (int k = 0; k < 6; ++k) {
        int d = lane + 32 * k;
        atomAddF(&op[d], al[d >> 6] * xs[d]);
    }
}

__global__ void k_leaky_inplace(float* __restrict__ a, int n, float slope)
{
    int i = blockIdx.x * blockDim.x + threadIdx.x;
    if (i < n) a[i] = lrelu(a[i], slope);
}

// out-logit = sum_s tanh(raw[r][s] + b[s]) * q[s]; optional transpose scatter
__global__ void k_rowdot_tanh(const float* __restrict__ raw, const float* __restrict__ b,
                              const float* __restrict__ q, float* __restrict__ out,
                              int R, int modN, int Mm)
{
    int r = blockIdx.x * blockDim.x + threadIdx.x;
    if (r >= R) return;
    const float* rp = raw + (size_t)r * 128;
    float acc = 0.f;
    for (int s = 0; s < 128; ++s) acc += tanhf(rp[s] + b[s]) * q[s];
    if (modN > 0) out[(r % modN) * Mm + r / modN] = acc;
    else          out[r] = acc;
}

// z[n][d] = sum_m out[m][n][d] * softmax_m(slog[n][m])
__global__ void k_sem_z(const float* __restrict__ outbuf, const float* __restrict__ slog,
                        float* __restrict__ z, int total)
{
    int i = blockIdx.x * blockDim.x + threadIdx.x;
    if (i >= total) return;
    int n = i / DD;
    float l0 = slog[n * 3 + 0], l1 = slog[n * 3 + 1], l2 = slog[n * 3 + 2];
    float mx = fmaxf(l0, fmaxf(l1, l2));
    float e0 = __expf(l0 - mx), e1 = __expf(l1 - mx), e2 = __expf(l2 - mx);
    float inv = 1.f / (e0 + e1 + e2);
    float acc = outbuf[i] * e0
              + outbuf[(size_t)N_PPI * DD + i] * e1
              + outbuf[2 * (size_t)N_PPI * DD + i] * e2;
    z[i] = acc * inv;
}

// softmax over 10000 nodes + attention pooling -> vec[192]
__global__ void k_pc_pool(const float* __restrict__ z, const float* __restrict__ pclog,
                          float* __restrict__ gam, float* __restrict__ vec)
{
    __shared__ float red[256];
    int t = threadIdx.x;
    float mx = -INFINITY;
    for (int n = t; n < N_PPI; n += 256) mx = fmaxf(mx, pclog[n]);
    red[t] = mx; __syncthreads();
    for (int s = 128; s > 0; s >>= 1) { if (t < s) red[t] = fmaxf(red[t], red[t + s]); __syncthreads(); }
    mx = red[0]; __syncthreads();
    float sm = 0.f;
    for (int n = t; n < N_PPI; n += 256) sm += __expf(pclog[n] - mx);
    red[t] = sm; __syncthreads();
    for (int s = 128; s > 0; s >>= 1) { if (t < s) red[t] += red[t + s]; __syncthreads(); }
    float inv = 1.f / red[0];
    __syncthreads();
    for (int n = t; n < N_PPI; n += 256) gam[n] = __expf(pclog[n] - mx) * inv;
    __syncthreads();
    for (int d = t; d < DD; d += 256) {
        float acc = 0.f;
        for (int n = 0; n < N_PPI; ++n) acc += z[(size_t)n * DD + d] * gam[n];
        vec[d] = acc;
    }
}

// ---------------------------------------------------------------------------
// Tiny metagraph sem_fuse (12 nodes, 2 metapaths, 40 edges): one block,
// global-scratch intermediates with __syncthreads phase barriers.
// ---------------------------------------------------------------------------
__global__ void k_mg_semfuse(const float* __restrict__ x, int K,
                             const float* __restrict__ Wl, const float* __restrict__ Wr,
                             const float* __restrict__ att, const float* __restrict__ bias,
                             const float* __restrict__ Wsem, const float* __restrict__ bsem,
                             const float* __restrict__ qsem,
                             const int* __restrict__ edges,   // [M_MG][2][E_MG]
                             float* __restrict__ outp,        // [12][192]
                             float* __restrict__ scr)
{
    const int NT = 256;
    const int ET = E_MG + N_MG;   // 52
    int t = threadIdx.x;
    float* xl = scr;                       // 12*192
    float* xr = xl + N_MG * DD;            // 12*192
    float* ab = xr + N_MG * DD;            // [M_MG][52][3]
    float* og = ab + M_MG * ET * 3;        // [M_MG][12][192]
    float* wt = og + M_MG * N_MG * DD;     // [M_MG*12][128]
    float* sl = wt + M_MG * N_MG * SEM;    // [12][M_MG]

    // 1: xl = x@Wl, xr = x@Wr
    for (int i = t; i < N_MG * DD; i += NT) {
        int n = i / DD, d = i % DD;
        float al = 0.f, ar = 0.f;
        for (int k = 0; k < K; ++k) {
            float xv = x[n * K + k];
            al += xv * Wl[k * DD + d];
            ar += xv * Wr[k * DD + d];
        }
        xl[i] = al; xr[i] = ar;
    }
    __syncthreads();
    // 2: per-edge logits
    for (int i = t; i < M_MG * ET * 3; i += NT) {
        int m = i / (ET * 3), r = i % (ET * 3), e = r / 3, h = r % 3;
        int src, dst;
        if (e < E_MG) { src = edges[(m * 2 + 0) * E_MG + e]; dst = edges[(m * 2 + 1) * E_MG + e]; }
        else          { src = dst = e - E_MG; }
        float acc = 0.f;
        for (int c = 0; c < OUT_CH; ++c) {
            float v = xl[src * DD + h * OUT_CH + c] + xr[dst * DD + h * OUT_CH + c];
            v = v > 0.f ? v : 0.2f * v;
            acc += v * att[h * OUT_CH + c];
        }
        ab[i] = acc;
    }
    __syncthreads();
    // 3: per-(m,dst,h) segment softmax, normalized alpha written in place
    for (int i = t; i < M_MG * N_MG * 3; i += NT) {
        int m = i / (N_MG * 3), r = i % (N_MG * 3), n = r / 3, h = r % 3;
        float mx = -INFINITY;
        for (int e = 0; e < ET; ++e) {
            int dst = (e < E_MG) ? edges[(m * 2 + 1) * E_MG + e] : e - E_MG;
            if (dst == n) mx = fmaxf(mx, ab[(m * ET + e) * 3 + h]);
        }
        float s = 0.f;
        for (int e = 0; e < ET; ++e) {
            int dst = (e < E_MG) ? edges[(m * 2 + 1) * E_MG + e] : e - E_MG;
            if (dst == n) s += __expf(ab[(m * ET + e) * 3 + h] - mx);
        }
        float inv = 1.f / (s + 1e-16f);
        for (int e = 0; e < ET; ++e) {
            int dst = (e < E_MG) ? edges[(m * 2 + 1) * E_MG + e] : e - E_MG;
            if (dst == n) {
                int id = (m * ET + e) * 3 + h;
                ab[id] = __expf(ab[id] - mx) * inv;
            }
        }
    }
    __syncthreads();
    // 4: aggregate + bias + leaky(0.01)
    for (int i = t; i < M_MG * N_MG * DD; i += NT) {
        int m = i / (N_MG * DD), r = i % (N_MG * DD), n = r / DD, d = r % DD;
        int h = d >> 6;
        float acc = bias[d];
        for (int e = 0; e < ET; ++e) {
            int src, dst;
            if (e < E_MG) { src = edges[(m * 2 + 0) * E_MG + e]; dst = edges[(m * 2 + 1) * E_MG + e]; }
            else          { src = dst = e - E_MG; }
            if (dst == n) acc += ab[(m * ET + e) * 3 + h] * xl[src * DD + d];
        }
        og[i] = lrelu(acc, 0.01f);
    }
    __syncthreads();
    // 5: wt = tanh(og @ Wsem + bsem)
    for (int i = t; i < M_MG * N_MG * SEM; i += NT) {
        int row = i / SEM, s = i % SEM;
        float acc = bsem[s];
        for (int d = 0; d < DD; ++d) acc += og[row * DD + d] * Wsem[d * SEM + s];
        wt[i] = tanhf(acc);
    }
    __syncthreads();
    // 6: semantic logits per (n,m)
    for (int i = t; i < N_MG * M_MG; i += NT) {
        int n = i / M_MG, m = i % M_MG;
        float acc = 0.f;
        for (int s = 0; s < SEM; ++s) acc += wt[(m * N_MG + n) * SEM + s] * qsem[s];
        sl[i] = acc;
    }
    __syncthreads();
    // 7: fuse
    for (int i = t; i < N_MG * DD; i += NT) {
        int n = i / DD, d = i % DD;
        float mx = -INFINITY;
        for (int m = 0; m < M_MG; ++m) mx = fmaxf(mx, sl[n * M_MG + m]);
        float es[M_MG], s = 0.f;
        for (int m = 0; m < M_MG; ++m) { es[m] = __expf(sl[n * M_MG + m] - mx); s += es[m]; }
        float acc = 0.f;
        for (int m = 0; m < M_MG; ++m) acc += og[(m * N_MG + n) * DD + d] * es[m];
        outp[i] = acc / s;
    }
}

// mg[:8] += vecs, then 100 x (4 sequential tissue neighbor-mean updates)
__global__ void k_mg_diffuse(float* __restrict__ mg, const float* __restrict__ vecs,
                             const int* __restrict__ tnb)
{
    int t = threadIdx.x;
    for (int i = t; i < C_TYPES * DD; i += 256) mg[i] += vecs[i];
    __syncthreads();
    for (int it = 0; it < TISSUE_UPDATE; ++it) {
        for (int ts = 0; ts < T_TIS; ++ts) {
            float v = 0.f;
            if (t < DD) {
                for (int k = 0; k < K_NB; ++k) v += mg[tnb[ts * K_NB + k] * DD + t];
                v *= (1.f / 3.f);
            }
            __syncthreads();
            if (t < DD) mg[(C_TYPES + ts) * DD + t] = v;
            __syncthreads();
        }
    }
}

// ---------------------------------------------------------------------------
extern "C" void kernel_launch(void* const* d_in, const int* in_sizes, int n_in,
                              void* d_out, int out_size, void* d_ws, size_t ws_size,
                              hipStream_t stream)
{
    const float* ppi_x    = (const float*)d_in[0];
    const float* mg_x     = (const float*)d_in[1];
    const int*   ppi_e    = (const int*)d_in[2];
    const int*   mg_e     = (const int*)d_in[3];
    const int*   tnb      = (const int*)d_in[4];
    const float* ppi_Wl   = (const float*)d_in[5];
    const float* ppi_Wr   = (const float*)d_in[6];
    const float* ppi_att  = (const float*)d_in[7];
    const float* ppi_b    = (const float*)d_in[8];
    const float* mgin_Wl  = (const float*)d_in[9];
    const float* mgin_Wr  = (const float*)d_in[10];
    const float* mgin_att = (const float*)d_in[11];
    const float* mgin_b   = (const float*)d_in[12];
    const float* mgout_Wl = (const float*)d_in[13];
    const float* mgout_Wr = (const float*)d_in[14];
    const float* mgout_att= (const float*)d_in[15];
    const float* mgout_b  = (const float*)d_in[16];
    const float* Wsem     = (const float*)d_in[17];
    const float* bsem     = (const float*)d_in[18];
    const float* qsem     = (const float*)d_in[19];
    const float* pcW      = (const float*)d_in[20];
    const float* pcb      = (const float*)d_in[21];
    const float* pcq      = (const float*)d_in[22];
    (void)in_sizes; (void)n_in; (void)out_size; (void)ws_size;

    float* out = (float*)d_out;

    // workspace carve-out (~62 MB, reused across the 8-celltype loop)
    char* ws = (char*)d_ws;
    size_t off = 0;
    auto alloc = [&](size_t bytes) -> char* {
        char* p = ws + off;
        off += (bytes + 255) & ~(size_t)255;
        return p;
    };
    float*    xl     = (float*)alloc(sizeof(float) * N_PPI * DD);
    float*    xr     = (float*)alloc(sizeof(float) * N_PPI * DD);
    float*    agg    = (float*)alloc(sizeof(float) * M_PPI * N_PPI * DD);
    float*    elog   = (float*)alloc(sizeof(float) * M_PPI * ETOT_PPI * 3);
    unsigned* mmax   = (unsigned*)alloc(sizeof(unsigned) * M_PPI * N_PPI * 3);
    float*    ssum   = (float*)alloc(sizeof(float) * M_PPI * N_PPI * 3);
    float*    semraw = (float*)alloc(sizeof(float) * M_PPI * N_PPI * SEM);
    float*    slog   = (float*)alloc(sizeof(float) * N_PPI * 3);
    float*    pcraw  = (float*)alloc(sizeof(float) * N_PPI * PC);
    float*    pclog  = (float*)alloc(sizeof(float) * N_PPI);
    float*    gam    = (float*)alloc(sizeof(float) * N_PPI);
    float*    vecs   = (float*)alloc(sizeof(float) * C_TYPES * DD);
    float*    mgbuf  = (float*)alloc(sizeof(float) * N_MG * DD);
    float*    mgscr  = (float*)alloc(sizeof(float) * 20000);

    // metagraph input conv (independent of PPI path)
    k_mg_semfuse<<<1, 256, 0, stream>>>(mg_x, IN_CH, mgin_Wl, mgin_Wr, mgin_att, mgin_b,
                                        Wsem, bsem, qsem, mg_e, mgbuf, mgscr);

    for (int c = 0; c < C_TYPES; ++c) {
        const float* xc   = ppi_x   + (size_t)c * N_PPI * IN_CH;
        const int*   ec   = ppi_e   + (size_t)c * M_PPI * 2 * E_PPI;
        const float* Wlc  = ppi_Wl  + (size_t)c * IN_CH * DD;
        const float* Wrc  = ppi_Wr  + (size_t)c * IN_CH * DD;
        const float* attc = ppi_att + (size_t)c * HEADS * OUT_CH;
        const float* bc   = ppi_b   + (size_t)c * DD;
        float*       zc   = out     + (size_t)c * N_PPI * DD;

        // xl = x@Wl, xr = x@Wr : [10000,128]@[128,192]  (WMMA + async-LDS B)
        {
            int mtiles = N_PPI / 16;                 // 625
            int blocks = (mtiles + 7) / 8;
            k_wmma_gemm_t<IN_CH, DD / 16><<<blocks, 256, 0, stream>>>(xc, Wlc, xl, mtiles);
            k_wmma_gemm_t<IN_CH, DD / 16><<<blocks, 256, 0, stream>>>(xc, Wrc, xr, mtiles);
        }
        // edge softmax + aggregation (3 metapaths, 60000 edges each)
        int nEdge = M_PPI * ETOT_PPI;
        k_init_edgebufs<<<(M_PPI * N_PPI * 3 + 255) / 256, 256, 0, stream>>>(mmax, ssum, M_PPI * N_PPI * 3);
        k_init_agg<<<(M_PPI * N_PPI * DD + 255) / 256, 256, 0, stream>>>(agg, bc, M_PPI * N_PPI * DD);
        k_edge_logits<<<(nEdge + 255) / 256, 256, 0, stream>>>(xl, xr, ec, attc, elog, mmax);
        k_edge_norm<<<(nEdge + 255) / 256, 256, 0, stream>>>(ec, elog, mmax, ssum);
        k_edge_agg<<<(nEdge * 32 + 255) / 256, 256, 0, stream>>>(xl, ec, elog, ssum, agg);
        k_leaky_inplace<<<(M_PPI * N_PPI * DD + 255) / 256, 256, 0, stream>>>(agg, M_PPI * N_PPI * DD, 0.01f);
        // semantic attention: [30000,192]@[192,128] (WMMA), tanh+q dot, fuse -> z (d_out)
        {
            int mtiles = (M_PPI * N_PPI) / 16;       // 1875
            k_wmma_gemm_t<DD, SEM / 16><<<(mtiles + 7) / 8, 256, 0, stream>>>(agg, Wsem, semraw, mtiles);
        }
        k_rowdot_tanh<<<(M_PPI * N_PPI + 255) / 256, 256, 0, stream>>>(semraw, bsem, qsem, slog,
                                                                       M_PPI * N_PPI, N_PPI, 3);
        k_sem_z<<<(N_PPI * DD + 255) / 256, 256, 0, stream>>>(agg, slog, zc, N_PPI * DD);
        // protein->celltype pooling: [10000,192]@[192,128] (WMMA), tanh+q, softmax pool
        {
            int mtiles = N_PPI / 16;                 // 625
            k_wmma_gemm_t<DD, PC / 16><<<(mtiles + 7) / 8, 256, 0, stream>>>(zc, pcW, pcraw, mtiles);
        }
        k_rowdot_tanh<<<(N_PPI + 255) / 256, 256, 0, stream>>>(pcraw, pcb, pcq, pclog, N_PPI, 0, 1);
        k_pc_pool<<<1, 256, 0, stream>>>(zc, pclog, gam, vecs + c * DD);
    }

    // mg[:8] += vecs; 100 rounds of sequential tissue neighbor means
    k_mg_diffuse<<<1, 256, 0, stream>>>(mgbuf, vecs, tnb);
    // metagraph output conv -> tail of d_out
    k_mg_semfuse<<<1, 256, 0, stream>>>(mgbuf, DD, mgout_Wl, mgout_Wr, mgout_att, mgout_b,
                                        Wsem, bsem, qsem, mg_e,
                                        out + (size_t)C_TYPES * N_PPI * DD, mgscr);
}